// GCN_73581379715089
// MI455X (gfx1250) — compile-verified
//
#include <hip/hip_runtime.h>

typedef __attribute__((ext_vector_type(2))) float v2f;
typedef __attribute__((ext_vector_type(8))) float v8f;

#define FIN 512
#define F1  16
#define F2  64

// ---------------- degree / symmetric normalization ----------------
__global__ void k_init_deg(float* __restrict__ deg, int n) {
  int i = blockIdx.x * blockDim.x + threadIdx.x;
  if (i < n) deg[i] = 1.0f;                    // self-loop contribution
}

__global__ void k_count_deg(const int* __restrict__ dst, float* __restrict__ deg, int ne) {
  int e = blockIdx.x * blockDim.x + threadIdx.x;
  if (e < ne) atomicAdd(&deg[dst[e]], 1.0f);
}

__global__ void k_dinv(float* __restrict__ deg, int n) {
  int i = blockIdx.x * blockDim.x + threadIdx.x;
  if (i < n) {
    float d = deg[i];
    deg[i] = (d > 0.0f) ? rsqrtf(d) : 0.0f;
  }
}

// ---------------- GEMM1: h[N,16] = x[N,512] @ W[512,16] (fp32 WMMA) ----------------
// One wave computes a 16x16 tile: 16 node-rows x 16 output features, K=512 in steps of 4.
__global__ void k_gemm1(const float* __restrict__ x, const float* __restrict__ W,
                        float* __restrict__ h, int n) {
  int wave = (blockIdx.x * blockDim.x + threadIdx.x) >> 5;
  int lane = threadIdx.x & 31;
  int row0 = wave * 16;
  if (row0 >= n) return;                        // wave-uniform exit: EXEC stays all-ones
  int m  = lane & 15;                           // row (A) / col (B)
  int hi = lane >> 4;                           // K-pair select per ISA 16x4 f32 A layout
  const float* arow = x + (size_t)(row0 + m) * FIN;
  v8f c = {};
  #pragma unroll 8
  for (int k0 = 0; k0 < FIN; k0 += 4) {
    int ka = k0 + 2 * hi;
    v2f a = *(const v2f*)(arow + ka);           // A[m][ka], A[m][ka+1]
    v2f b;
    b.x = W[(size_t)ka * F1 + m];               // B[ka][m]
    b.y = W[(size_t)(ka + 1) * F1 + m];         // B[ka+1][m]
    c = __builtin_amdgcn_wmma_f32_16x16x4_f32(false, a, false, b, (short)0, c, false, false);
  }
  float* hrow = h + (size_t)row0 * F1;
  #pragma unroll
  for (int j = 0; j < 8; ++j)                   // C/D layout: VGPR j -> M = j + 8*hi, N = m
    hrow[(size_t)(j + 8 * hi) * F1 + m] = c[j];
}

// ---------------- GEMM2: h2[N,64] = h1[N,16] @ W2[16,64] (fp32 WMMA) ----------------
// One wave per 16x16 tile; 4 column tiles cover the 64 output features. K=16 -> 4 WMMAs.
__global__ void k_gemm2(const float* __restrict__ h1, const float* __restrict__ W,
                        float* __restrict__ h2, int n) {
  int wave = (blockIdx.x * blockDim.x + threadIdx.x) >> 5;
  int lane = threadIdx.x & 31;
  int rt = wave >> 2;
  int ct = wave & 3;
  int row0 = rt * 16;
  if (row0 >= n) return;
  int m  = lane & 15;
  int hi = lane >> 4;
  const float* arow = h1 + (size_t)(row0 + m) * F1;
  v8f c = {};
  #pragma unroll
  for (int k0 = 0; k0 < F1; k0 += 4) {
    int ka = k0 + 2 * hi;
    v2f a = *(const v2f*)(arow + ka);
    v2f b;
    b.x = W[(size_t)ka * F2 + ct * 16 + m];
    b.y = W[(size_t)(ka + 1) * F2 + ct * 16 + m];
    c = __builtin_amdgcn_wmma_f32_16x16x4_f32(false, a, false, b, (short)0, c, false, false);
  }
  #pragma unroll
  for (int j = 0; j < 8; ++j)
    h2[(size_t)(row0 + j + 8 * hi) * F2 + ct * 16 + m] = c[j];
}

// ---------------- self-loop init: agg[n][f] = h[n][f] * dinv[n]^2 ----------------
// float4-vectorized; compile-time F so node index is a shift. Also initializes the accumulator (no separate zeroing).
template <int F>
__global__ void k_selfinit(const float4* __restrict__ hp4, const float* __restrict__ dinv,
                           float4* __restrict__ agg4, int n) {
  constexpr int G = F / 4;
  long long idx = (long long)blockIdx.x * blockDim.x + threadIdx.x;
  if (idx >= (long long)n * G) return;
  int node = (int)(idx / G);
  float di = dinv[node];
  float s = di * di;
  float4 v = hp4[idx];
  v.x *= s; v.y *= s; v.z *= s; v.w *= s;
  agg4[idx] = v;
}

// ---------------- edge scatter: agg[d] += h[s] * dinv[s]*dinv[d] ----------------
// TPE threads per edge; each thread owns a contiguous 16-float (4 x float4) chunk:
// indices + dinv loaded once per 16 features, b128 gathers, fp32 global atomics.
template <int F, int TPE>
__global__ void k_scatter(const int* __restrict__ src, const int* __restrict__ dst,
                          const float* __restrict__ hp, const float* __restrict__ dinv,
                          float* __restrict__ agg, int ne) {
  constexpr int CH = F / TPE;                 // floats per thread (16)
  long long idx = (long long)blockIdx.x * blockDim.x + threadIdx.x;
  if (idx >= (long long)ne * TPE) return;
  int e = (int)(idx / TPE);                   // TPE is a power of two -> shift
  int q = (int)(idx % TPE);
  int s = src[e], d = dst[e];
  const float* vp = hp + (size_t)s * F + q * CH;
  __builtin_prefetch(vp, 0, 0);               // global_prefetch: overlap row fetch with dinv loads
  float w = dinv[s] * dinv[d];
  float* op = agg + (size_t)d * F + q * CH;
  #pragma unroll
  for (int g = 0; g < CH / 4; ++g) {
    float4 v = *(const float4*)(vp + 4 * g);
    atomicAdd(op + 4 * g + 0, v.x * w);
    atomicAdd(op + 4 * g + 1, v.y * w);
    atomicAdd(op + 4 * g + 2, v.z * w);
    atomicAdd(op + 4 * g + 3, v.w * w);
  }
}

// ---------------- bias + relu (layer 1), bias only (layer 2), float4-vectorized ----------------
template <int F>
__global__ void k_relu_bias(float4* __restrict__ agg4, const float4* __restrict__ b4, int n) {
  constexpr int G = F / 4;
  long long idx = (long long)blockIdx.x * blockDim.x + threadIdx.x;
  if (idx >= (long long)n * G) return;
  float4 b = b4[(int)(idx % G)];
  float4 v = agg4[idx];
  v.x += b.x; v.y += b.y; v.z += b.z; v.w += b.w;
  v.x = v.x > 0.0f ? v.x : 0.0f;
  v.y = v.y > 0.0f ? v.y : 0.0f;
  v.z = v.z > 0.0f ? v.z : 0.0f;
  v.w = v.w > 0.0f ? v.w : 0.0f;
  agg4[idx] = v;
}

template <int F>
__global__ void k_bias(float4* __restrict__ out4, const float4* __restrict__ b4, int n) {
  constexpr int G = F / 4;
  long long idx = (long long)blockIdx.x * blockDim.x + threadIdx.x;
  if (idx >= (long long)n * G) return;
  float4 b = b4[(int)(idx % G)];
  float4 v = out4[idx];
  v.x += b.x; v.y += b.y; v.z += b.z; v.w += b.w;
  out4[idx] = v;
}

static inline int cdiv_ll(long long a, long long b) { return (int)((a + b - 1) / b); }

extern "C" void kernel_launch(void* const* d_in, const int* in_sizes, int n_in,
                              void* d_out, int out_size, void* d_ws, size_t ws_size,
                              hipStream_t stream) {
  const float* x  = (const float*)d_in[0];
  const int*   ei = (const int*)d_in[1];
  const float* W1 = (const float*)d_in[2];
  const float* b1 = (const float*)d_in[3];
  const float* W2 = (const float*)d_in[4];
  const float* b2 = (const float*)d_in[5];
  float* out = (float*)d_out;

  const int N = in_sizes[0] / FIN;     // 100000
  const int E = in_sizes[1] / 2;       // 3200000
  const int* src = ei;
  const int* dst = ei + E;

  // workspace layout (floats): dinv[N] | h1_pre[16N] | agg1/h1relu[16N] | h2_pre[64N]
  float* wsf  = (float*)d_ws;
  float* dinv = wsf;
  float* h1p  = wsf + (size_t)N;
  float* agg1 = wsf + (size_t)17 * N;
  float* h2p  = wsf + (size_t)33 * N;

  const int B = 256;                   // 8 wave32s per block

  // normalization: deg -> dinv
  k_init_deg <<<cdiv_ll(N, B), B, 0, stream>>>(dinv, N);
  k_count_deg<<<cdiv_ll(E, B), B, 0, stream>>>(dst, dinv, E);
  k_dinv     <<<cdiv_ll(N, B), B, 0, stream>>>(dinv, N);

  // layer 1
  long long waves1 = (long long)((N + 15) / 16);
  k_gemm1<<<cdiv_ll(waves1 * 32, B), B, 0, stream>>>(x, W1, h1p, N);
  k_selfinit<F1><<<cdiv_ll((long long)N * (F1 / 4), B), B, 0, stream>>>(
      (const float4*)h1p, dinv, (float4*)agg1, N);
  k_scatter<F1, 1><<<cdiv_ll((long long)E * 1, B), B, 0, stream>>>(
      src, dst, h1p, dinv, agg1, E);
  k_relu_bias<F1><<<cdiv_ll((long long)N * (F1 / 4), B), B, 0, stream>>>(
      (float4*)agg1, (const float4*)b1, N);

  // layer 2 (accumulate straight into d_out)
  long long waves2 = waves1 * 4;
  k_gemm2<<<cdiv_ll(waves2 * 32, B), B, 0, stream>>>(agg1, W2, h2p, N);
  k_selfinit<F2><<<cdiv_ll((long long)N * (F2 / 4), B), B, 0, stream>>>(
      (const float4*)h2p, dinv, (float4*)out, N);
  k_scatter<F2, 4><<<cdiv_ll((long long)E * 4, B), B, 0, stream>>>(
      src, dst, h2p, dinv, out, E);
  k_bias<F2><<<cdiv_ll((long long)N * (F2 / 4), B), B, 0, stream>>>(
      (float4*)out, (const float4*)b2, N);
}